// LayerStacks_6262062317931
// MI455X (gfx1250) — compile-verified
//
#include <hip/hip_runtime.h>

// LayerStacks: x[B,1024] -> per-sample stack selection (8 stacks) ->
//   [1024->16 clip01] -> [16->32 clip01] -> [32->1]
// Strategy: bucket samples by stack idx, then WMMA f32 16x16x4 GEMM per
// 16-sample same-stack tile (K=1024), fused stage2/3 in VALU.

#define L1X2   1024
#define NL2    16
#define NL3    32
#define NSTACK 8
#define W1PAD  1028   // 16-row LDS stride; 1028 mod 64 = 4 -> conflict-free B reads
#define Y1PAD  17
#define BPS    64     // blocks per stack
#define WAVES  8      // 256-thread blocks, wave32

typedef __attribute__((ext_vector_type(2))) float v2f;
typedef __attribute__((ext_vector_type(8))) float v8f;

__global__ void ls_init(int* cnt) {
    if (threadIdx.x < 2 * NSTACK) cnt[threadIdx.x >= NSTACK ? 8 + threadIdx.x : threadIdx.x] = 0;
}
// ws layout (ints): [0..7]=counts [8..15]=offsets [16..23]=cursors [24..31]=pad [32..]=bucket

__global__ void ls_hist(const int* __restrict__ idx, int* __restrict__ counts, int B) {
    int s = blockIdx.x * blockDim.x + threadIdx.x;
    if (s < B) atomicAdd(&counts[idx[s]], 1);
}

__global__ void ls_scan(const int* __restrict__ counts, int* __restrict__ offsets) {
    int acc = 0;
    for (int c = 0; c < NSTACK; ++c) { offsets[c] = acc; acc += counts[c]; }
}

__global__ void ls_scatter(const int* __restrict__ idx, const int* __restrict__ offsets,
                           int* __restrict__ cursors, int* __restrict__ bucket, int B) {
    int s = blockIdx.x * blockDim.x + threadIdx.x;
    if (s < B) {
        int c = idx[s];
        int pos = offsets[c] + atomicAdd(&cursors[c], 1);
        bucket[pos] = s;
    }
}

__global__ __launch_bounds__(256)
void ls_main(const float* __restrict__ x,
             const float* __restrict__ w1, const float* __restrict__ b1,
             const float* __restrict__ w2, const float* __restrict__ b2,
             const float* __restrict__ wo, const float* __restrict__ bo,
             const int* __restrict__ counts, const int* __restrict__ offsets,
             const int* __restrict__ bucket, float* __restrict__ out) {
    __shared__ float w1s[NL2 * W1PAD];      // padded stage-1 weights of this stack
    __shared__ float w2s[NL3 * NL2];
    __shared__ float b1s[NL2];
    __shared__ float b2s[NL3];
    __shared__ float wos[NL3];
    __shared__ float y1s[WAVES][NL2 * Y1PAD];  // per-wave stage-1 transpose buffer

    const int c   = blockIdx.y;
    const int tid = threadIdx.x;

    for (int i = tid; i < NL2 * L1X2; i += 256) {
        int r = i >> 10, col = i & 1023;
        w1s[r * W1PAD + col] = w1[(c * NL2 + r) * L1X2 + col];
    }
    for (int i = tid; i < NL3 * NL2; i += 256) {
        int r = i >> 4, col = i & 15;
        w2s[i] = w2[(c * NL3 + r) * NL2 + col];
    }
    if (tid < NL2) b1s[tid] = b1[c * NL2 + tid];
    if (tid < NL3) { b2s[tid] = b2[c * NL3 + tid]; wos[tid] = wo[c * NL3 + tid]; }
    __syncthreads();

    const int n_c = counts[c];
    if (n_c == 0) return;
    const int   base = offsets[c];
    const float bo_c = bo[c];
    const int nTiles = (n_c + 15) >> 4;

    const int wave = tid >> 5;
    const int lane = tid & 31;
    const int ml   = lane & 15;   // A row (sample) / B col (neuron) / D col
    const int half = lane >> 4;   // K sub-pair select
    const int mlr  = ml * W1PAD;

    for (int t = blockIdx.x * WAVES + wave; t < nTiles; t += gridDim.x * WAVES) {
        // --- stage 1: D[16 samp x 16 neuron] = X[16x1024] * W1c^T, f32 WMMA ---
        int gs   = t * 16 + ml;
        int cgs  = gs < n_c ? gs : n_c - 1;           // pad partial tile (masked store)
        int samp = bucket[base + cgs];
        const float* xrow = x + (long)samp * L1X2;

        v8f acc = {0.f, 0.f, 0.f, 0.f, 0.f, 0.f, 0.f, 0.f};
        #pragma unroll 8
        for (int k = 0; k < L1X2; k += 4) {
            const int ko = k + 2 * half;
            v2f a = __builtin_nontemporal_load((const v2f*)(xrow + ko)); // stream x, keep L2 for weights
            v2f b = *(const v2f*)(w1s + mlr + ko);
            acc = __builtin_amdgcn_wmma_f32_16x16x4_f32(
                false, a, false, b, (short)0, acc, false, false);
        }

        // bias + clip01, transpose D layout -> row-major via per-wave LDS
        float* yb = &y1s[wave][0];
        #pragma unroll
        for (int r = 0; r < 8; ++r) {
            float v = acc[r] + b1s[ml];
            v = fminf(fmaxf(v, 0.f), 1.f);
            yb[(8 * half + r) * Y1PAD + ml] = v;      // sample row = 8*half+r, neuron = ml
        }
        asm volatile("" ::: "memory");  // wave-local LDS is in-order; keep compiler honest

        // --- stage 2 + 3: lane = (sample ml, jj half). 16x16 MAC + clip + wo dot ---
        float xv[NL2];
        #pragma unroll
        for (int j = 0; j < NL2; ++j) xv[j] = yb[ml * Y1PAD + j];

        float acc3 = 0.f;
        #pragma unroll
        for (int q = 0; q < 16; ++q) {
            int jj = half * 16 + q;
            float s2 = b2s[jj];
            #pragma unroll
            for (int j = 0; j < NL2; ++j) s2 += xv[j] * w2s[jj * NL2 + j];
            s2 = fminf(fmaxf(s2, 0.f), 1.f);
            acc3 += s2 * wos[jj];
        }
        acc3 += __shfl_xor(acc3, 16, 32);             // combine jj halves

        if (half == 0 && gs < n_c) out[bucket[base + gs]] = acc3 + bo_c;
    }
}

extern "C" void kernel_launch(void* const* d_in, const int* in_sizes, int n_in,
                              void* d_out, int out_size, void* d_ws, size_t ws_size,
                              hipStream_t stream) {
    const float* x   = (const float*)d_in[0];
    const int*   idx = (const int*)  d_in[1];
    const float* w1  = (const float*)d_in[2];
    const float* b1  = (const float*)d_in[3];
    const float* w2  = (const float*)d_in[4];
    const float* b2  = (const float*)d_in[5];
    const float* wo  = (const float*)d_in[6];
    const float* bo  = (const float*)d_in[7];
    float* out = (float*)d_out;

    const int B = in_sizes[1];
    int* wsi     = (int*)d_ws;
    int* counts  = wsi;
    int* offsets = wsi + 8;
    int* cursors = wsi + 16;
    int* bucket  = wsi + 32;

    ls_init<<<1, 32, 0, stream>>>(counts);           // zeroes counts + cursors
    int nb = (B + 255) / 256;
    ls_hist<<<nb, 256, 0, stream>>>(idx, counts, B);
    ls_scan<<<1, 1, 0, stream>>>(counts, offsets);
    ls_scatter<<<nb, 256, 0, stream>>>(idx, offsets, cursors, bucket, B);

    dim3 grid(BPS, NSTACK, 1);
    ls_main<<<grid, 256, 0, stream>>>(x, w1, b1, w2, b2, wo, bo,
                                      counts, offsets, bucket, out);
}